// SimpleGCN_60902636257457
// MI455X (gfx1250) — compile-verified
//
#include <hip/hip_runtime.h>
#include <hip/hip_bf16.h>

// ---------------------------------------------------------------------------
// SimpleGCN on MI455X (gfx1250).
//   4 x { GEMM[100000x128 @ 128x128] -> self-loop+bias -> edge scatter } ->
//   per-graph mean pool -> 128x10 MLP.
// GEMM uses V_WMMA_F32_16X16X4_F32 (full fp32 precision; GEMM is not the
// bottleneck, the L2-resident edge scatter is, so fp32 WMMA is the right
// precision choice against an fp32 reference).
// ---------------------------------------------------------------------------

#define GCN_N      100000
#define GCN_E      1600000
#define GCN_F      128
#define GCN_G      64
#define GCN_C      10

typedef float v2f __attribute__((ext_vector_type(2)));
typedef float v8f __attribute__((ext_vector_type(8)));

// LDS padding to avoid bank conflicts
#define WPAD 132   // W rows padded: 128 -> 132 floats
#define APAD 130   // A rows padded: 128 -> 130 floats (even -> b64 aligned)
#define GEMM_LDS_FLOATS (128 * WPAD + 16 * APAD)

// ---------------- degree / dinv ----------------

__global__ __launch_bounds__(256) void k_count_deg(
    const int* __restrict__ dst, int* __restrict__ cnt, int n_edges) {
  int e = blockIdx.x * blockDim.x + threadIdx.x;
  if (e < n_edges) atomicAdd(&cnt[dst[e]], 1);
}

__global__ __launch_bounds__(256) void k_dinv(
    const int* __restrict__ cnt, float* __restrict__ dinv, int n_nodes) {
  int i = blockIdx.x * blockDim.x + threadIdx.x;
  if (i < n_nodes) {
    float deg = (float)cnt[i] + 1.0f;   // +1 for the self loop
    dinv[i] = 1.0f / sqrtf(deg);
  }
}

// ---------------- dense GEMM: T = relu?(H) @ W ----------------
// Block: 256 threads = 8 wave32.  Each block: 16 rows x all 128 cols.
// Wave w computes the 16x16 tile at columns [16w, 16w+16) with 32 chained
// v_wmma_f32_16x16x4_f32 ops.
//
// Fragment layouts (ISA 7.12.2):
//   A (16x4 f32):  lane L -> row m = L&15; v0 = A[m][k0 + 2*(L>>4)],
//                                          v1 = A[m][k0 + 2*(L>>4) + 1]
//   B (4x16 f32):  lane L -> col n = L&15; v0 = B[k0 + 2*(L>>4)][n],
//                                          v1 = B[k0 + 2*(L>>4) + 1][n]
//   C/D (16x16):   VGPR r, lane L -> D[r + 8*(L>>4)][L&15]

__global__ __launch_bounds__(256) void k_gemm_wmma(
    const float* __restrict__ H, const float* __restrict__ W,
    float* __restrict__ T, int relu_in) {
  extern __shared__ float smem[];
  float* sW = smem;                 // 128 x WPAD
  float* sA = smem + 128 * WPAD;    // 16 x APAD

  const int tid  = threadIdx.x;
  const int wave = tid >> 5;
  const int lane = tid & 31;
  const int row0 = blockIdx.x * 16;

  // Stage W (128x128) into LDS, row-padded.
  for (int i = tid; i < 128 * 128; i += 256) {
    int r = i >> 7, c = i & 127;
    sW[r * WPAD + c] = W[i];
  }
  // Stage the 16x128 A slab, folding in ReLU of the previous layer.
  for (int i = tid; i < 16 * 128; i += 256) {
    int r = i >> 7, c = i & 127;
    float v = H[(size_t)(row0 + r) * 128 + c];
    if (relu_in) v = fmaxf(v, 0.0f);
    sA[r * APAD + c] = v;
  }
  __syncthreads();

  const int n0 = wave * 16;
  const int m  = lane & 15;          // A row
  const int n  = n0 + (lane & 15);   // B/D column
  const int kk = (lane >> 4) * 2;    // lane-half K offset

  v8f c = {};
#pragma unroll
  for (int k0 = 0; k0 < 128; k0 += 4) {
    v2f a = *(const v2f*)&sA[m * APAD + k0 + kk];        // ds_load_b64
    v2f b;
    b.x = sW[(k0 + kk)     * WPAD + n];
    b.y = sW[(k0 + kk + 1) * WPAD + n];
    c = __builtin_amdgcn_wmma_f32_16x16x4_f32(
        /*neg_a=*/false, a, /*neg_b=*/false, b,
        /*c_mod=*/(short)0, c, /*reuse_a=*/false, /*reuse_b=*/false);
  }

  const int mbase = (lane >> 4) * 8;
#pragma unroll
  for (int r = 0; r < 8; ++r)
    T[(size_t)(row0 + mbase + r) * 128 + n] = c[r];
}

// ---------------- agg init: self-loop term + bias ----------------
// agg[i][f] = dinv[i]^2 * T[i][f] + b[f]

__global__ __launch_bounds__(256) void k_init_agg(
    const float* __restrict__ T, const float* __restrict__ dinv,
    const float* __restrict__ bias, float* __restrict__ agg, int total) {
  int i = blockIdx.x * blockDim.x + threadIdx.x;
  if (i < total) {
    int node = i >> 7, f = i & 127;
    float d = dinv[node];
    agg[i] = d * d * T[i] + bias[f];
  }
}

// ---------------- edge scatter: agg[dst] += norm * T[src] ----------------
// One wave32 per edge; each lane moves a float4 (128 floats / 32 lanes).

__global__ __launch_bounds__(256) void k_scatter(
    const int* __restrict__ src, const int* __restrict__ dst,
    const float* __restrict__ dinv, const float* __restrict__ T,
    float* __restrict__ agg, int n_edges) {
  int e    = (int)((blockIdx.x * (size_t)blockDim.x + threadIdx.x) >> 5);
  int lane = threadIdx.x & 31;
  if (e >= n_edges) return;
  int s = src[e];
  int d = dst[e];
  float norm = dinv[s] * dinv[d];
  const float4 v = ((const float4*)(T + (size_t)s * 128))[lane];
  float* a = agg + (size_t)d * 128 + lane * 4;
  unsafeAtomicAdd(a + 0, v.x * norm);
  unsafeAtomicAdd(a + 1, v.y * norm);
  unsafeAtomicAdd(a + 2, v.z * norm);
  unsafeAtomicAdd(a + 3, v.w * norm);
}

// ---------------- per-graph counts ----------------

__global__ __launch_bounds__(256) void k_graph_count(
    const int* __restrict__ batch, int* __restrict__ gcnt, int n_nodes) {
  int i = blockIdx.x * blockDim.x + threadIdx.x;
  if (i < n_nodes) atomicAdd(&gcnt[batch[i]], 1);
}

// ---------------- pooled sums of relu(agg), exploiting sorted batch -------
// 128 threads (one per feature) walk 256 nodes, register-accumulate per
// graph, flush with one atomic per graph boundary.

#define POOL_NODES 256

__global__ __launch_bounds__(128) void k_pool_sum(
    const float* __restrict__ agg, const int* __restrict__ batch,
    float* __restrict__ pooled, int n_nodes) {
  int f = threadIdx.x;  // 0..127
  int start = blockIdx.x * POOL_NODES;
  if (start >= n_nodes) return;
  int end = min(start + POOL_NODES, n_nodes);
  int curg = batch[start];
  float acc = 0.0f;
  for (int i = start; i < end; ++i) {
    int g = batch[i];
    if (g != curg) {
      unsafeAtomicAdd(&pooled[(size_t)curg * 128 + f], acc);
      acc = 0.0f;
      curg = g;
    }
    acc += fmaxf(agg[(size_t)i * 128 + f], 0.0f);
  }
  unsafeAtomicAdd(&pooled[(size_t)curg * 128 + f], acc);
}

// ---------------- final MLP: out[g][c] = mean(pooled[g]) @ W_mlp + b ------

__global__ __launch_bounds__(640) void k_mlp(
    const float* __restrict__ pooled, const int* __restrict__ gcnt,
    const float* __restrict__ Wm, const float* __restrict__ bm,
    float* __restrict__ out) {
  int t = threadIdx.x;
  if (t >= GCN_G * GCN_C) return;
  int g = t / GCN_C, c = t % GCN_C;
  float inv = 1.0f / fmaxf((float)gcnt[g], 1.0f);
  float acc = bm[c];
  for (int f = 0; f < 128; ++f)
    acc += pooled[(size_t)g * 128 + f] * inv * Wm[f * GCN_C + c];
  out[(size_t)g * GCN_C + c] = acc;
}

// ---------------------------------------------------------------------------

extern "C" void kernel_launch(void* const* d_in, const int* in_sizes, int n_in,
                              void* d_out, int out_size, void* d_ws, size_t ws_size,
                              hipStream_t stream) {
  const float* x      = (const float*)d_in[0];   // [N,128]
  const int*   eidx   = (const int*)d_in[1];     // [2,E] row-major
  const int*   batch  = (const int*)d_in[2];     // [N]
  const float* W_in   = (const float*)d_in[3];   // [128,128]
  const float* b_in   = (const float*)d_in[4];   // [128]
  const float* W_mid  = (const float*)d_in[5];   // [128,128]
  const float* b_mid  = (const float*)d_in[6];   // [128]
  const float* W_mlp  = (const float*)d_in[7];   // [128,10]
  const float* b_mlp  = (const float*)d_in[8];   // [10]
  float*       out    = (float*)d_out;           // [64,10]

  const int N = in_sizes[0] / GCN_F;
  const int E = in_sizes[1] / 2;
  const int* src = eidx;
  const int* dst = eidx + E;

  // Workspace layout (bytes)
  char* ws = (char*)d_ws;
  const size_t MB = 1024 * 1024;
  int*   cnt    = (int*)  (ws + 0);         // N ints
  float* dinv   = (float*)(ws + 1 * MB);    // N floats
  float* tbuf   = (float*)(ws + 2 * MB);    // N*128 floats (51.2 MB)
  float* agg    = (float*)(ws + 56 * MB);   // N*128 floats (51.2 MB)
  float* pooled = (float*)(ws + 108 * MB);  // 64*128 floats
  int*   gcnt   = (int*)  (ws + 109 * MB);  // 64 ints

  const size_t NF = (size_t)N * GCN_F;
  const int gemm_blocks  = N / 16;                      // 6250
  const int gemm_shmem   = GEMM_LDS_FLOATS * sizeof(float);
  const int nf_blocks    = (int)((NF + 255) / 256);
  const int edge_blocks1 = (E + 255) / 256;             // 1 thread/edge
  const int edge_blocks8 = (E + 7) / 8;                 // 1 wave/edge
  const int node_blocks  = (N + 255) / 256;
  const int pool_blocks  = (N + POOL_NODES - 1) / POOL_NODES;

  // --- degree / dinv ---
  hipMemsetAsync(cnt, 0, (size_t)N * sizeof(int), stream);
  k_count_deg<<<edge_blocks1, 256, 0, stream>>>(dst, cnt, E);
  k_dinv<<<node_blocks, 256, 0, stream>>>(cnt, dinv, N);

  // --- layer 1: x @ W_in ---
  k_gemm_wmma<<<gemm_blocks, 256, gemm_shmem, stream>>>(x, W_in, tbuf, 0);
  k_init_agg<<<nf_blocks, 256, 0, stream>>>(tbuf, dinv, b_in, agg, (int)NF);
  k_scatter<<<edge_blocks8, 256, 0, stream>>>(src, dst, dinv, tbuf, agg, E);

  // --- layers 2..4: relu(agg) @ W_mid (shared weights) ---
  for (int l = 0; l < 3; ++l) {
    k_gemm_wmma<<<gemm_blocks, 256, gemm_shmem, stream>>>(agg, W_mid, tbuf, 1);
    k_init_agg<<<nf_blocks, 256, 0, stream>>>(tbuf, dinv, b_mid, agg, (int)NF);
    k_scatter<<<edge_blocks8, 256, 0, stream>>>(src, dst, dinv, tbuf, agg, E);
  }

  // --- pooling + MLP ---
  hipMemsetAsync(pooled, 0, (size_t)GCN_G * GCN_F * sizeof(float), stream);
  hipMemsetAsync(gcnt, 0, (size_t)GCN_G * sizeof(int), stream);
  k_graph_count<<<node_blocks, 256, 0, stream>>>(batch, gcnt, N);
  k_pool_sum<<<pool_blocks, 128, 0, stream>>>(agg, batch, pooled, N);
  k_mlp<<<1, 640, 0, stream>>>(pooled, gcnt, W_mlp, b_mlp, out);
}